// CTCLayer_18322330485233
// MI455X (gfx1250) — compile-verified
//
#include <hip/hip_runtime.h>
#include <math.h>

typedef float v2f __attribute__((ext_vector_type(2)));
typedef float v8f __attribute__((ext_vector_type(8)));

constexpr int   B_    = 64;
constexpr int   T_    = 2048;
constexpr int   C_    = 512;
constexpr int   L_    = 256;
constexpr int   S_    = 2 * L_ + 1;   // 513
constexpr int   BLANK = C_ - 1;       // 511
constexpr float EPS_  = 1e-7f;
constexpr float NEGI  = -1e30f;

// ---------------------------------------------------------------------------
// Phase 1: logdenom[r] = log( sum_c y[r,c] + C*eps )  for r in [0, B*T)
// Row sums done on the matrix pipe: D = A(16x4,f32) x ones(4x16) accumulated
// over K=512 in 128 V_WMMA_F32_16X16X4_F32 steps (exact f32).
// A layout (32-bit 16x4): lanes 0-15 hold row M, VGPR0=K0,VGPR1=K1;
// lanes 16-31 hold K2,K3.  => lane loads float2 at col 4k + 2*(lane>>4).
// D layout: VGPR j, lanes 0-15 -> row j ; lanes 16-31 -> row j+8.
// ---------------------------------------------------------------------------
__global__ __launch_bounds__(256) void ctc_rowsum_wmma(const float* __restrict__ y,
                                                       float* __restrict__ logdenom)
{
    const int wave    = (blockIdx.x * 256 + threadIdx.x) >> 5;  // 0..8191
    const int lane    = threadIdx.x & 31;
    const int rowbase = wave * 16;

    const int r    = lane & 15;
    const int koff = (lane >> 4) * 2;
    const float* p = y + (size_t)(rowbase + r) * C_ + koff;

    v8f c = {0.f, 0.f, 0.f, 0.f, 0.f, 0.f, 0.f, 0.f};
    v2f ones; ones.x = 1.0f; ones.y = 1.0f;

    #pragma unroll 8
    for (int k = 0; k < 128; ++k) {
        v2f a = *(const v2f*)(p + 4 * k);
        c = __builtin_amdgcn_wmma_f32_16x16x4_f32(false, a, false, ones,
                                                  (short)0, c, false, false);
    }

    // lanes 0 and 16 each write 8 row results (compile-time vector indices)
    if ((lane & 15) == 0) {
        const int ro = rowbase + (lane >> 4) * 8;
        #pragma unroll
        for (int j = 0; j < 8; ++j)
            logdenom[ro + j] = __logf(c[j] + (float)C_ * EPS_);
    }
}

// ---------------------------------------------------------------------------
// Phase 2: one block per batch. 544 threads (17 waves), state s = tid < 513.
// Depth-4 async global->LDS row pipeline (waves 0-3 issue B128, 2KB/row).
// Per step: s_wait_asynccnt<=2 (row t landed), barrier, issue row t+3 into
// its (now dead) buffer, then the log-sum-exp shift update.
// ---------------------------------------------------------------------------
__global__ __launch_bounds__(544) void ctc_alpha(const int* __restrict__ y_true,
                                                 const float* __restrict__ y_pred,
                                                 const float* __restrict__ logdenom,
                                                 float* __restrict__ out)
{
    const int b   = blockIdx.x;
    const int tid = threadIdx.x;

    __shared__ float rowbuf[4][C_];     // 8 KB quad-buffered probability row
    __shared__ float alpha[2][S_ + 3];  // ping-pong alpha
    __shared__ float ldno[T_];          // 8 KB per-batch log denominators
    __shared__ int   exts[S_];
    __shared__ int   allow[S_];

    const float* ybase = y_pred + (size_t)b * T_ * C_;

    if (tid < S_)
        exts[tid] = (tid & 1) ? y_true[b * L_ + (tid >> 1)] : BLANK;
    for (int i = tid; i < T_; i += 544)
        ldno[i] = logdenom[b * T_ + i];
    __syncthreads();
    if (tid < S_) {
        const int e = exts[tid];
        allow[tid] = (tid >= 2) && (e != BLANK) && (e != exts[tid - 2]);
    }

    // prologue: issue rows 0,1,2
    if (tid < 128) {
        #pragma unroll
        for (int t = 0; t < 3; ++t) {
            unsigned lds = (unsigned)(size_t)&rowbuf[t][tid * 4];
            unsigned long long ga =
                (unsigned long long)(const void*)(ybase + (size_t)t * C_ + tid * 4);
            asm volatile("global_load_async_to_lds_b128 %0, %1, off"
                         :: "v"(lds), "v"(ga) : "memory");
        }
        asm volatile("s_wait_asynccnt 0x2" ::: "memory");  // row 0 landed
    }
    __syncthreads();
    if (tid < 128) {  // issue row 3
        unsigned lds = (unsigned)(size_t)&rowbuf[3][tid * 4];
        unsigned long long ga =
            (unsigned long long)(const void*)(ybase + (size_t)3 * C_ + tid * 4);
        asm volatile("global_load_async_to_lds_b128 %0, %1, off"
                     :: "v"(lds), "v"(ga) : "memory");
    }
    // alpha init at t=0
    if (tid < S_) {
        float v = NEGI;
        if (tid < 2) v = __logf(rowbuf[0][exts[tid]] + EPS_) - ldno[0];
        alpha[0][tid] = v;
    }

    for (int t = 1; t < T_; ++t) {
        if (tid < 128)
            asm volatile("s_wait_asynccnt 0x2" ::: "memory");  // row t landed
        __syncthreads();  // alpha[t-1] visible; rowbuf[(t+3)&3] no longer read
        if (tid < 128) {
            const int tf = t + 3;
            const int tc = (tf < T_) ? tf : (T_ - 1);  // clamp keeps cnt pattern uniform
            unsigned lds = (unsigned)(size_t)&rowbuf[tf & 3][tid * 4];
            unsigned long long ga =
                (unsigned long long)(const void*)(ybase + (size_t)tc * C_ + tid * 4);
            asm volatile("global_load_async_to_lds_b128 %0, %1, off"
                         :: "v"(lds), "v"(ga) : "memory");
        }
        if (tid < S_) {
            const float* ap = alpha[(t - 1) & 1];
            const float a0 = ap[tid];
            const float a1 = (tid >= 1) ? ap[tid - 1] : NEGI;
            const float a2 = (allow[tid]) ? ap[tid - 2] : NEGI;
            const float m  = fmaxf(a0, fmaxf(a1, a2));
            const float comb =
                m + __logf(__expf(a0 - m) + __expf(a1 - m) + __expf(a2 - m));
            const float lp = __logf(rowbuf[t & 3][exts[tid]] + EPS_) - ldno[t];
            alpha[t & 1][tid] = comb + lp;
        }
    }
    __syncthreads();
    if (tid == 0) {
        const float aN  = alpha[(T_ - 1) & 1][S_ - 1];
        const float aN1 = alpha[(T_ - 1) & 1][S_ - 2];
        const float m   = fmaxf(aN, aN1);
        out[b] = -(m + __logf(__expf(aN - m) + __expf(aN1 - m)));
    }
}

extern "C" void kernel_launch(void* const* d_in, const int* in_sizes, int n_in,
                              void* d_out, int out_size, void* d_ws, size_t ws_size,
                              hipStream_t stream) {
    const int*   y_true = (const int*)d_in[0];
    const float* y_pred = (const float*)d_in[1];
    float*       out    = (float*)d_out;
    float*       logden = (float*)d_ws;   // needs B*T*4 = 512 KB of scratch

    // 131072 rows, 16 rows per wave, 8 waves per block -> 1024 blocks
    ctc_rowsum_wmma<<<1024, 256, 0, stream>>>(y_pred, logden);
    ctc_alpha<<<B_, 544, 0, stream>>>(y_true, y_pred, logden, out);
}